// NTKSVGP_50637664419912
// MI455X (gfx1250) — compile-verified
//
#include <hip/hip_runtime.h>

// NTKSVGP on MI455X (gfx1250). FP32 throughout (Cholesky + 1e-3 jitter needs it);
// Gram/GEMM stages use V_WMMA_F32_16X16X4_F32. Total work ~1 GFLOP, ~10 MB
// workspace -> fully L2-resident, latency-bound.

typedef __attribute__((ext_vector_type(2))) float v2f;
typedef __attribute__((ext_vector_type(8))) float v8f;

#define Dd 128
#define Hh 128
#define Cc 10
#define Nn 256
#define Mm 128
#define Tt 128
#define STOT 512            // N + M + T concatenated sample count
#define SCALE (1.0f/256.0f) // 1/(DELTA*N)
#define JIT 1e-3f
#define INV_SIG2 1.0f       // 1/SIGMA2

// One k-step (K=4) of a Gram tile: O[m,n] += sum_{kk<4} FL[m,k+kk]*FR[n,k+kk]
// A (16x4 MxK) f32 layout: lane l<16 -> {K0,K1} of row m0+l; lane l+16 -> {K2,K3}.
// B (4x16 KxN) f32 layout assumed: VGPR0 = {row K0 | row K1}, VGPR1 = {row K2 | row K3}.
__device__ __forceinline__ v8f gram4(v8f acc, const float4 a4, const float4 b4, int half) {
  v2f A, B;
  A[0] = half ? a4.z : a4.x;
  A[1] = half ? a4.w : a4.y;
  B[0] = half ? b4.y : b4.x;
  B[1] = half ? b4.w : b4.z;
  return __builtin_amdgcn_wmma_f32_16x16x4_f32(false, A, false, B, (short)0, acc,
                                               false, false);
}

// ---------------------------------------------------------------- stage A ----
// Per-sample features: h1, h2, and per-class g1_c, g2_c (the NTK factorization).
__global__ void ntk_features(const float* __restrict__ X, const float* __restrict__ Z,
                             const float* __restrict__ Xt,
                             const float* __restrict__ W1, const float* __restrict__ b1,
                             const float* __restrict__ W2, const float* __restrict__ b2,
                             const float* __restrict__ W3,
                             float* __restrict__ Xcat, float* __restrict__ H1,
                             float* __restrict__ H2, float* __restrict__ G1,
                             float* __restrict__ G2) {
  __shared__ float xs[Dd], h1s[Hh], g2s[Hh];
  const int s = blockIdx.x, tid = threadIdx.x;
  const float* xin = (s < Nn) ? (X + s * Dd)
                              : ((s < Nn + Mm) ? (Z + (s - Nn) * Dd)
                                               : (Xt + (s - Nn - Mm) * Dd));
  float xv = xin[tid];
  xs[tid] = xv;
  Xcat[s * Dd + tid] = xv;
  __syncthreads();

  float a1 = b1[tid];
  for (int d = 0; d < Dd; ++d) a1 = fmaf(xs[d], W1[d * Hh + tid], a1);
  float h1 = tanhf(a1);
  h1s[tid] = h1;
  H1[s * Hh + tid] = h1;
  __syncthreads();

  __builtin_prefetch(W2 + tid * Hh, 0, 3);  // rows reused in the g1 matvec below
  float a2 = b2[tid];
  for (int d = 0; d < Hh; ++d) a2 = fmaf(h1s[d], W2[d * Hh + tid], a2);
  float h2 = tanhf(a2);
  H2[s * Hh + tid] = h2;
  __syncthreads();

  const float t1 = 1.0f - h1 * h1, t2 = 1.0f - h2 * h2;
  for (int c = 0; c < Cc; ++c) {
    float g2 = W3[tid * Cc + c] * t2;         // W3 is (H,C) row-major
    g2s[tid] = g2;
    G2[((size_t)c * STOT + s) * Hh + tid] = g2;
    __syncthreads();
    float acc = 0.f;
    for (int j = 0; j < Hh; ++j) acc = fmaf(W2[tid * Hh + j], g2s[j], acc);
    G1[((size_t)c * STOT + s) * Hh + tid] = t1 * acc;
    __syncthreads();
  }
}

// ---------------------------------------------------------------- stage B ----
// Fused NTK tiles: Kall[c][m in Z][n in X|Z|T], 5 fp32 WMMA Gram accumulators.
__global__ void ntk_gram(const float* __restrict__ Xcat, const float* __restrict__ H1,
                         const float* __restrict__ H2, const float* __restrict__ G1,
                         const float* __restrict__ G2, float* __restrict__ Kall) {
  const int lane = threadIdx.x, l = lane & 15, half = lane >> 4;
  const int n0 = blockIdx.x * 16, m0 = blockIdx.y * 16, c = blockIdx.z;
  const int zs = Nn + m0 + l;  // FL sample (inducing point Z)
  const int cs = n0 + l;       // FR sample (concat column)
  const float* pLx = Xcat + zs * Dd;  const float* pRx = Xcat + cs * Dd;
  const float* pL1 = H1 + zs * Hh;    const float* pR1 = H1 + cs * Hh;
  const float* pL2 = H2 + zs * Hh;    const float* pR2 = H2 + cs * Hh;
  const float* pLg1 = G1 + ((size_t)c * STOT + zs) * Hh;
  const float* pRg1 = G1 + ((size_t)c * STOT + cs) * Hh;
  const float* pLg2 = G2 + ((size_t)c * STOT + zs) * Hh;
  const float* pRg2 = G2 + ((size_t)c * STOT + cs) * Hh;

  v8f sx = {}, s1 = {}, s2 = {}, sg1 = {}, sg2 = {};
  for (int k = 0; k < Dd; k += 4) {
    sx  = gram4(sx,  *(const float4*)(pLx + k),  *(const float4*)(pRx + k),  half);
    s1  = gram4(s1,  *(const float4*)(pL1 + k),  *(const float4*)(pR1 + k),  half);
    s2  = gram4(s2,  *(const float4*)(pL2 + k),  *(const float4*)(pR2 + k),  half);
    sg1 = gram4(sg1, *(const float4*)(pLg1 + k), *(const float4*)(pRg1 + k), half);
    sg2 = gram4(sg2, *(const float4*)(pLg2 + k), *(const float4*)(pRg2 + k), half);
  }
  // C/D layout: VGPR v holds rows v (lanes 0-15) and v+8 (lanes 16-31).
  for (int v = 0; v < 8; ++v) {
    const int row = m0 + half * 8 + v, col = n0 + l;
    const float kv =
        SCALE * ((sx[v] + 1.f) * sg1[v] + (s1[v] + 1.f) * sg2[v] + s2[v] + 1.f);
    Kall[((size_t)c * Mm + row) * STOT + col] = kv;
  }
}

// Kxx diagonal for test points (self-dots only).
__global__ void ntk_diag(const float* __restrict__ Xcat, const float* __restrict__ H1,
                         const float* __restrict__ H2, const float* __restrict__ G1,
                         const float* __restrict__ G2, float* __restrict__ Kdiag) {
  const int t = threadIdx.x;
  const int s = Nn + Mm + t;
  const float* px = Xcat + s * Dd;
  const float* p1 = H1 + s * Hh;
  const float* p2 = H2 + s * Hh;
  float sx = 0.f, s1 = 0.f, s2 = 0.f;
  for (int k = 0; k < Dd; ++k) {
    sx = fmaf(px[k], px[k], sx);
    s1 = fmaf(p1[k], p1[k], s1);
    s2 = fmaf(p2[k], p2[k], s2);
  }
  for (int c = 0; c < Cc; ++c) {
    const float* pg1 = G1 + ((size_t)c * STOT + s) * Hh;
    const float* pg2 = G2 + ((size_t)c * STOT + s) * Hh;
    float a = 0.f, b = 0.f;
    for (int k = 0; k < Hh; ++k) {
      a = fmaf(pg1[k], pg1[k], a);
      b = fmaf(pg2[k], pg2[k], b);
    }
    Kdiag[c * Tt + t] = SCALE * ((sx + 1.f) * a + (s1 + 1.f) * b + s2 + 1.f);
  }
}

// ---------------------------------------------------------------- stage C ----
// A0 = Kzz + j*I (-> Lm), A1 = Kzz + beta_u + 2j*I (-> Lb); beta_u via WMMA Gram.
__global__ void ntk_build_ab(const float* __restrict__ Kall, float* __restrict__ A0,
                             float* __restrict__ A1) {
  const int lane = threadIdx.x, l = lane & 15, half = lane >> 4;
  const int k0 = blockIdx.x * 16, m0 = blockIdx.y * 16, c = blockIdx.z;
  const float* base = Kall + (size_t)c * Mm * STOT;
  const float* pL = base + (size_t)(m0 + l) * STOT;  // Kzx row m
  const float* pR = base + (size_t)(k0 + l) * STOT;  // Kzx row k
  v8f acc = {};
  for (int n = 0; n < Nn; n += 4)
    acc = gram4(acc, *(const float4*)(pL + n), *(const float4*)(pR + n), half);
  for (int v = 0; v < 8; ++v) {
    const int row = m0 + half * 8 + v, col = k0 + l;
    const float kzz = base[(size_t)row * STOT + Nn + col];  // Kzz[row][col]
    const float d = (row == col) ? JIT : 0.f;
    const size_t o = ((size_t)c * Mm + row) * Mm + col;
    A0[o] = kzz + d;
    A1[o] = kzz + INV_SIG2 * acc[v] + 2.f * d;
  }
}

// In-LDS Cholesky of the 20 SPD 128x128 matrices (blocks 0..9 -> A0, 10..19 -> A1).
__global__ void ntk_chol(float* __restrict__ A0, float* __restrict__ A1) {
  __shared__ float As[Mm * Mm];  // 64 KB of the 320 KB WGP LDS
  const int mat = blockIdx.x, tid = threadIdx.x;
  float* A = (mat < Cc) ? (A0 + (size_t)mat * Mm * Mm)
                        : (A1 + (size_t)(mat - Cc) * Mm * Mm);
  for (int j = 0; j < Mm; ++j) As[tid * Mm + j] = A[tid * Mm + j];
  __syncthreads();
  for (int k = 0; k < Mm; ++k) {
    if (tid == k) As[k * Mm + k] = sqrtf(As[k * Mm + k]);
    __syncthreads();
    if (tid > k) As[tid * Mm + k] /= As[k * Mm + k];
    __syncthreads();
    const float lik = (tid > k) ? As[tid * Mm + k] : 0.f;
    for (int j = k + 1; j < Mm; ++j) {
      const float ljk = As[j * Mm + k];
      if (tid >= j) As[tid * Mm + j] = fmaf(-lik, ljk, As[tid * Mm + j]);
    }
    __syncthreads();
  }
  for (int j = 0; j < Mm; ++j) A[tid * Mm + j] = As[tid * Mm + j];
}

// Lambda_u, alpha_u solve (via Lb), f_mean, and triangular-solve variance terms.
// Note: Lam = alpha + F*beta = Y/sigma^2, so F and b3 cancel exactly.
__global__ void ntk_solve(const float* __restrict__ Kall, const float* __restrict__ A0,
                          const float* __restrict__ A1, const float* __restrict__ Y,
                          const float* __restrict__ Kdiag, float* __restrict__ out) {
  __shared__ float bvec[Mm], alph[Mm];
  const int tid = threadIdx.x, c = blockIdx.x;
  const float* Kc = Kall + (size_t)c * Mm * STOT;
  const float* Lm = A0 + (size_t)c * Mm * Mm;
  const float* Lb = A1 + (size_t)c * Mm * Mm;

  {  // Lambda_u[m] = sum_n Kzx[m,n] * Y[n,c]/sigma^2
    float acc = 0.f;
    for (int n = 0; n < Nn; ++n) acc = fmaf(Kc[(size_t)tid * STOT + n], Y[n * Cc + c], acc);
    bvec[tid] = acc * INV_SIG2;
  }
  __syncthreads();
  for (int j = 0; j < Mm; ++j) {  // forward: Lb y = b
    if (tid == j) bvec[j] /= Lb[j * Mm + j];
    __syncthreads();
    const float yj = bvec[j];
    if (tid > j) bvec[tid] = fmaf(-Lb[tid * Mm + j], yj, bvec[tid]);
    __syncthreads();
  }
  for (int j = Mm - 1; j >= 0; --j) {  // backward: Lb^T a = y
    if (tid == j) bvec[j] /= Lb[j * Mm + j];
    __syncthreads();
    const float xj = bvec[j];
    if (tid < j) bvec[tid] = fmaf(-Lb[j * Mm + tid], xj, bvec[tid]);
    __syncthreads();
  }
  alph[tid] = bvec[tid];
  __syncthreads();

  const int t = tid;
  float fm = 0.f;  // f_mean[t,c] = sum_m Kxz[t,m] * alpha_u[m]
  for (int m = 0; m < Mm; ++m)
    fm = fmaf(Kc[(size_t)m * STOT + Nn + Mm + t], alph[m], fm);
  out[t * Cc + c] = fm;

  float xcol[Mm];  // Am(:,t) = Lm^{-1} Kzxt(:,t), then Ab with Lb
  float am2 = 0.f;
  for (int j = 0; j < Mm; ++j) {
    float acc = Kc[(size_t)j * STOT + Nn + Mm + t];
    for (int i = 0; i < j; ++i) acc = fmaf(-Lm[j * Mm + i], xcol[i], acc);
    const float v = acc / Lm[j * Mm + j];
    xcol[j] = v;
    am2 = fmaf(v, v, am2);
  }
  float ab2 = 0.f;
  for (int j = 0; j < Mm; ++j) {
    float acc = Kc[(size_t)j * STOT + Nn + Mm + t];
    for (int i = 0; i < j; ++i) acc = fmaf(-Lb[j * Mm + i], xcol[i], acc);
    const float v = acc / Lb[j * Mm + j];
    xcol[j] = v;
    ab2 = fmaf(v, v, ab2);
  }
  out[Tt * Cc + t * Cc + c] = Kdiag[c * Tt + t] - am2 + ab2;
}

extern "C" void kernel_launch(void* const* d_in, const int* in_sizes, int n_in,
                              void* d_out, int out_size, void* d_ws, size_t ws_size,
                              hipStream_t stream) {
  const float* X  = (const float*)d_in[0];
  const float* Y  = (const float*)d_in[1];
  const float* Z  = (const float*)d_in[2];
  const float* Xt = (const float*)d_in[3];
  const float* W1 = (const float*)d_in[4];
  const float* b1 = (const float*)d_in[5];
  const float* W2 = (const float*)d_in[6];
  const float* b2 = (const float*)d_in[7];
  const float* W3 = (const float*)d_in[8];
  // d_in[9] = b3: cancels analytically (Lam = Y/sigma^2, NTK is b3-independent).
  (void)in_sizes; (void)n_in; (void)out_size;

  float* ws = (float*)d_ws;
  size_t o = 0;
  float* Xcat = ws + o; o += (size_t)STOT * Dd;       //  256 KB
  float* H1   = ws + o; o += (size_t)STOT * Hh;       //  256 KB
  float* H2   = ws + o; o += (size_t)STOT * Hh;       //  256 KB
  float* G1   = ws + o; o += (size_t)Cc * STOT * Hh;  //  2.5 MB
  float* G2   = ws + o; o += (size_t)Cc * STOT * Hh;  //  2.5 MB
  float* Kall = ws + o; o += (size_t)Cc * Mm * STOT;  //  2.5 MB
  float* A0   = ws + o; o += (size_t)Cc * Mm * Mm;    //  640 KB
  float* A1   = ws + o; o += (size_t)Cc * Mm * Mm;    //  640 KB
  float* Kdg  = ws + o; o += (size_t)Cc * Tt;         //    5 KB
  (void)ws_size;  // ~9.6 MB total

  float* out = (float*)d_out;

  ntk_features<<<STOT, 128, 0, stream>>>(X, Z, Xt, W1, b1, W2, b2, W3,
                                         Xcat, H1, H2, G1, G2);
  ntk_gram<<<dim3(STOT / 16, Mm / 16, Cc), 32, 0, stream>>>(Xcat, H1, H2, G1, G2, Kall);
  ntk_diag<<<1, Tt, 0, stream>>>(Xcat, H1, H2, G1, G2, Kdg);
  ntk_build_ab<<<dim3(Mm / 16, Mm / 16, Cc), 32, 0, stream>>>(Kall, A0, A1);
  ntk_chol<<<2 * Cc, 128, 0, stream>>>(A0, A1);
  ntk_solve<<<Cc, 128, 0, stream>>>(Kall, A0, A1, Y, Kdg, out);
}